// BehlerG2_73976516706437
// MI455X (gfx1250) — compile-verified
//
#include <hip/hip_runtime.h>
#include <math.h>

typedef __attribute__((ext_vector_type(2))) float v2f;
typedef __attribute__((ext_vector_type(8))) float v8f;

#define RC_CUT 5.0f
#define AA 192      // atoms
#define TT 1536     // triples per atom
#define NN 64       // PBC offsets per atom
#define EE 8        // etas
#define ZZ 4        // zetas (-> 8 angular channels)
#define WPB 4       // waves per block (wave32)

// Per-wave LDS ordering: same-wave DS ops are processed in-order (ISA §7.3),
// s_wait_dscnt 0 guarantees all prior LDS stores completed; "memory" clobber
// stops the compiler from caching/reordering shared-memory accesses around it.
#define WAVE_LDS_FENCE() asm volatile("s_wait_dscnt 0" ::: "memory")

// base^n for n in [0,15], branch-free (select-product over precomputed powers).
// Correct for negative bases (1 - cos_t <= 0 here). Powers are hoisted by caller.
__device__ __forceinline__ float ipow_sel(float b1, float b2, float b4, float b8, int n) {
  float r = (n & 1) ? b1 : 1.0f;
  r *= (n & 2) ? b2 : 1.0f;
  r *= (n & 4) ? b4 : 1.0f;
  r *= (n & 8) ? b8 : 1.0f;
  return r;
}

__global__ __launch_bounds__(WPB * 32)
void behler_angular_wmma(const float* __restrict__ pos,    // (B,A,3)
                         const float* __restrict__ cell,   // (B,3,3)
                         const float* __restrict__ offs,   // (B,A,N,3)
                         const float* __restrict__ etas,   // (E)
                         const int*   __restrict__ zetas,  // (ZZ)
                         const int*   __restrict__ nbrj,   // (B,A,T)
                         const int*   __restrict__ nbrk,
                         const int*   __restrict__ offj,
                         const int*   __restrict__ offk,
                         const int*   __restrict__ maskt,
                         float*       __restrict__ out,    // (B,A,64)
                         int BA) {
  // per-wave LDS staging: cart offset table + radius/ang tiles (stride 9 => conflict-free)
  __shared__ float s_cart[WPB][NN][3];
  __shared__ float s_rad[WPB][32][9];
  __shared__ float s_ang[WPB][32][9];

  const int lane = threadIdx.x & 31;
  const int wv   = threadIdx.x >> 5;
  const int ba   = blockIdx.x * WPB + wv;
  if (ba >= BA) return;              // no block barriers below; per-wave exit is safe
  const int b = ba / AA;
  const int a = ba - b * AA;

  // cell rows -> registers; cart[n][c] = sum_d offs[b,a,n,d] * cell[b,d,c]
  float cl[9];
#pragma unroll
  for (int i = 0; i < 9; ++i) cl[i] = cell[b * 9 + i];
  for (int n = lane; n < NN; n += 32) {
    const float* o = offs + (((size_t)b * AA + a) * NN + n) * 3;
    const float d0 = o[0], d1 = o[1], d2 = o[2];
    s_cart[wv][n][0] = d0 * cl[0] + d1 * cl[3] + d2 * cl[6];
    s_cart[wv][n][1] = d0 * cl[1] + d1 * cl[4] + d2 * cl[7];
    s_cart[wv][n][2] = d0 * cl[2] + d1 * cl[5] + d2 * cl[8];
  }

  float et[EE];
#pragma unroll
  for (int e = 0; e < EE; ++e) et[e] = etas[e];

  float slo[ZZ], shi[ZZ];
  int zv[ZZ];
#pragma unroll
  for (int z = 0; z < ZZ; ++z) {
    zv[z]  = zetas[z];
    slo[z] = ldexpf(1.0f, 1 - zv[z]);   // 2^(1-zeta)
    shi[z] = ldexpf(1.0f, 1 + zv[z]);   // 2^(1+zeta)
  }

  const float rix = pos[((size_t)b * AA + a) * 3 + 0];
  const float riy = pos[((size_t)b * AA + a) * 3 + 1];
  const float riz = pos[((size_t)b * AA + a) * 3 + 2];

  WAVE_LDS_FENCE();   // s_cart visible to this wave's readers

  v8f acc = {0.f, 0.f, 0.f, 0.f, 0.f, 0.f, 0.f, 0.f};
  const size_t triBase = (size_t)ba * TT;
  const float kpi = 3.14159265358979f / RC_CUT;

  for (int t0 = 0; t0 < TT; t0 += 32) {
    const size_t ti = triBase + t0 + lane;

    // prefetch next chunk's index streams (global_prefetch_b8) to hide gather latency
    if (t0 + 32 < TT) {
      const size_t tn = ti + 32;
      __builtin_prefetch(&nbrj[tn], 0, 1);
      __builtin_prefetch(&nbrk[tn], 0, 1);
      __builtin_prefetch(&offj[tn], 0, 1);
      __builtin_prefetch(&offk[tn], 0, 1);
      __builtin_prefetch(&maskt[tn], 0, 1);
    }

    const int j  = nbrj[ti];
    const int k  = nbrk[ti];
    const int oj = offj[ti];
    const int ok = offk[ti];
    const float m = (float)maskt[ti];

    const float* pj = pos + ((size_t)b * AA + j) * 3;
    const float* pk = pos + ((size_t)b * AA + k) * 3;
    const float pjx = pj[0] + s_cart[wv][oj][0];
    const float pjy = pj[1] + s_cart[wv][oj][1];
    const float pjz = pj[2] + s_cart[wv][oj][2];
    const float pkx = pk[0] + s_cart[wv][ok][0];
    const float pky = pk[1] + s_cart[wv][ok][1];
    const float pkz = pk[2] + s_cart[wv][ok][2];

    const float djx = pjx - rix, djy = pjy - riy, djz = pjz - riz;
    const float dkx = pkx - rix, dky = pky - riy, dkz = pkz - riz;
    const float ejx = pkx - pjx, ejy = pky - pjy, ejz = pkz - pjz;

    const float r2ij = fmaxf(djx * djx + djy * djy + djz * djz, 1e-12f);
    const float r2ik = fmaxf(dkx * dkx + dky * dky + dkz * dkz, 1e-12f);
    const float r2jk = fmaxf(ejx * ejx + ejy * ejy + ejz * ejz, 1e-12f);
    const float rij = sqrtf(r2ij), rik = sqrtf(r2ik), rjk = sqrtf(r2jk);
    const float r2 = r2ij + r2ik + r2jk;

    const float fij = (rij < RC_CUT) ? 0.5f * (cosf(kpi * rij) + 1.0f) : 0.0f;
    const float fik = (rik < RC_CUT) ? 0.5f * (cosf(kpi * rik) + 1.0f) : 0.0f;
    const float fjk = (rjk < RC_CUT) ? 0.5f * (cosf(kpi * rjk) + 1.0f) : 0.0f;
    const float cutm = fij * fik * fjk * m;   // fold mask into radius factor

    const float cost = (m != 0.0f) ? (r2 / (2.0f * rij * rik)) : 0.0f;
    const float b1 = 1.0f - cost;
    const float b2 = b1 * b1;
    const float b4 = b2 * b2;
    const float b8 = b4 * b4;

#pragma unroll
    for (int e = 0; e < EE; ++e)
      s_rad[wv][lane][e] = expf(-et[e] * r2) * cutm;
#pragma unroll
    for (int z = 0; z < ZZ; ++z) {
      const float p = ipow_sel(b1, b2, b4, b8, zv[z]);
      s_ang[wv][lane][z]      = slo[z] * p;
      s_ang[wv][lane][ZZ + z] = shi[z] * p;
    }

    WAVE_LDS_FENCE();   // staging tile complete before re-layout reads

    // A 16x4 f32 layout: lane = M row; VGPR0 holds K0 (lanes 0-15) / K2 (lanes 16-31)
    // B  4x16 f32 layout: lane = N col; VGPR0 holds K0 row (lanes 0-15) / K2 row (lanes 16-31)
    const int half = (lane < 16) ? 0 : 2;
    const int row  = lane & 15;
#pragma unroll
    for (int g = 0; g < 8; ++g) {
      const int tA = g * 4 + half;
      v2f av, bv;
      av.x = (row < 8) ? s_rad[wv][tA][row]     : 0.0f;
      av.y = (row < 8) ? s_rad[wv][tA + 1][row] : 0.0f;
      bv.x = (row < 8) ? s_ang[wv][tA][row]     : 0.0f;
      bv.y = (row < 8) ? s_ang[wv][tA + 1][row] : 0.0f;
      acc = __builtin_amdgcn_wmma_f32_16x16x4_f32(
          /*neg_a=*/false, av, /*neg_b=*/false, bv,
          /*c_mod=*/(short)0, acc, /*reuse_a=*/false, /*reuse_b=*/false);
    }

    WAVE_LDS_FENCE();   // readback done before next chunk overwrites the tile
  }

  // D layout: VGPR r, lanes 0-15 hold M=r, N=lane  ->  f[e=r][z=lane] for lane<8
  if (lane < 8) {
    float* o = out + (size_t)ba * 64;
#pragma unroll
    for (int r = 0; r < 8; ++r) o[r * 8 + lane] = acc[r];
  }
}

extern "C" void kernel_launch(void* const* d_in, const int* in_sizes, int n_in,
                              void* d_out, int out_size, void* d_ws, size_t ws_size,
                              hipStream_t stream) {
  const float* positions = (const float*)d_in[0];
  const float* cell      = (const float*)d_in[1];
  const float* offsets   = (const float*)d_in[2];
  const float* etas      = (const float*)d_in[3];
  const int*   zetas     = (const int*)d_in[4];
  const int*   nbrj      = (const int*)d_in[5];
  const int*   nbrk      = (const int*)d_in[6];
  const int*   offj      = (const int*)d_in[7];
  const int*   offk      = (const int*)d_in[8];
  const int*   maskt     = (const int*)d_in[9];
  float*       out       = (float*)d_out;

  const int B  = in_sizes[0] / (AA * 3);
  const int BA = B * AA;
  const int grid = (BA + WPB - 1) / WPB;

  behler_angular_wmma<<<grid, WPB * 32, 0, stream>>>(
      positions, cell, offsets, etas, zetas, nbrj, nbrk, offj, offk, maskt, out, BA);
}